// Cycler_90555090469566
// MI455X (gfx1250) — compile-verified
//
#include <hip/hip_runtime.h>
#include <math.h>

#define N_IN 10000
#define MN   10000
#define EE   320000
#define DD   64
#define HH   256

typedef float v2f __attribute__((ext_vector_type(2)));
typedef float v8f __attribute__((ext_vector_type(8)));
typedef unsigned int v4u __attribute__((ext_vector_type(4)));
typedef int v4i __attribute__((ext_vector_type(4)));
typedef int v8i __attribute__((ext_vector_type(8)));

__device__ __forceinline__ float lk(float x){ return x >= 0.f ? x : 0.2f*x; }
// order-preserving float->uint key for atomic max (0 == below -inf since self-loops cover all dst)
__device__ __forceinline__ unsigned fkey(float f){ unsigned u=__float_as_uint(f); return (u&0x80000000u)? ~u : (u|0x80000000u); }
__device__ __forceinline__ float fdec(unsigned k){ unsigned u=(k&0x80000000u)? (k&0x7fffffffu) : ~k; return __uint_as_float(u); }

// --- TDM: async-load one [16 m rows][kc k cols] fp32 tile of W into LDS -------
// D# per CDNA5 ISA ch.8: 2D tensor at gaddr, dim0 = kc (OOB k reads -> 0),
// dim0_stride = N_IN, tile 64x16, LDS padding 4 dwords per 64 -> row pitch 68.
#define LDS_PITCH 68
__device__ __forceinline__ void tdm_load_wtile(const float* gaddr, unsigned lds_off, int kc)
{
  unsigned long long ga = (unsigned long long)(size_t)gaddr;
  v4u g0;
  g0[0] = 1u;                                            // count=1, user mode, no gather
  g0[1] = lds_off;                                       // LDS byte address
  g0[2] = (unsigned)(ga & 0xffffffffu);                  // global_addr[31:0]
  g0[3] = (unsigned)((ga >> 32) & 0x01ffffffu)           // global_addr[56:32]
          | 0x80000000u;                                 // type=2 ("image")
  v8i g1;
  g1[0] = (2 << 16)                                      // data_size = 4B
        | (1 << 20)                                      // pad_enable
        | (5 << 22)                                      // pad_interval = 64 dwords
        | (3 << 25);                                     // pad_amount  = 4 dwords
  g1[1] = (int)(((unsigned)kc & 0xffffu) << 16);         // tensor_dim0[15:0]
  g1[2] = (int)(16u << 16);                              // tensor_dim0 hi=0 | tensor_dim1=16
  g1[3] = (int)(64u << 16);                              // tensor_dim1 hi=0 | tile_dim0=64
  g1[4] = 16;                                            // tile_dim1=16, tile_dim2=0
  g1[5] = N_IN;                                          // tensor_dim0_stride[31:0]
  g1[6] = 0;                                             // stride hi | dim1_stride lo
  g1[7] = 0;                                             // dim1_stride hi (unused, 2D)
  v4i z4 = {0, 0, 0, 0};
  v8i z8 = {0, 0, 0, 0, 0, 0, 0, 0};
  __builtin_amdgcn_tensor_load_to_lds(g0, g1, z4, z4, z8, 0);  // clang-23: 6-arg form
}

// ---------------- GEMM1: t[64][M] = x^T(64 x N) * W_trs^T(N x M), fp32 WMMA ----
// block = 128 thr (4 waves) -> one 16-wide m tile; wave w -> d rows [16w,16w+16).
// W tiles streamed by the Tensor Data Mover, double-buffered in LDS.
__global__ __launch_bounds__(128) void k_gemm1(const float* __restrict__ x,
                                               const float* __restrict__ W,
                                               float* __restrict__ t)
{
  __shared__ float wl[2][16 * LDS_PITCH];  // 2 x 4352 B, TDM-padded row pitch 68
  const int tid  = threadIdx.x;
  const int wave = tid >> 5, lane = tid & 31;
  const int lo = lane & 15, hi = lane >> 4;
  const int mbase = blockIdx.x * 16;
  const int dbase = wave * 16;
  const unsigned lds0 = (unsigned)(size_t)(void*)&wl[0][0];
  const unsigned lds1 = (unsigned)(size_t)(void*)&wl[1][0];
  const float* wrow = W + (size_t)mbase * N_IN;
  const int nch = (N_IN + 63) >> 6;        // 157 chunks of K=64 (tail zero-filled)

  v8f c = {};
  if (wave == 0) tdm_load_wtile(wrow, lds0, min(64, N_IN));
  for (int i = 0; i < nch; ++i) {
    if (wave == 0) __builtin_amdgcn_s_wait_tensorcnt(0);  // chunk i landed in LDS
    __syncthreads();                                      // ...and prev compute done
    if (wave == 0 && i + 1 < nch) {                       // prefetch chunk i+1 via TDM
      int k0n = (i + 1) << 6;
      tdm_load_wtile(wrow + k0n, (i & 1) ? lds0 : lds1, min(64, N_IN - k0n));
    }
    const float* wc = &wl[i & 1][0];
    const int k0 = i << 6;
    #pragma unroll 4
    for (int s = 0; s < 64; s += 4) {
      const int k = k0 + s;
      // A 16x4: lane=row(d), VGPR0 K=2*hi, VGPR1 K=2*hi+1 (clamp: B is 0 past K)
      const int ka = min(k + 2*hi,     N_IN - 1);
      const int kb = min(k + 2*hi + 1, N_IN - 1);
      v2f a, b;
      a.x = x[(size_t)ka * DD + dbase + lo];
      a.y = x[(size_t)kb * DD + dbase + lo];
      // B 4x16: lane=col(m); pitch-68 rows => banks (4*lo+s+2*hi)%64, conflict-free
      b.x = wc[lo * LDS_PITCH + s + 2*hi];
      b.y = wc[lo * LDS_PITCH + s + 2*hi + 1];
      c = __builtin_amdgcn_wmma_f32_16x16x4_f32(false, a, false, b, (short)0, c, false, false);
    }
  }
  #pragma unroll
  for (int r = 0; r < 8; ++r) {            // C: VGPR r = rows r (lanes 0-15) / r+8 (16-31)
    int d = dbase + r + 8*hi;
    t[(size_t)d * MN + mbase + lo] = c[r]; // b_trs cancelled by BN mean-subtract
  }
}

// ------------- stage B: BN2(axis0) + leaky + transpose + adain1, fully fused ----
__global__ __launch_bounds__(256) void k_stageB(const float* __restrict__ t,
    const float* __restrict__ g, const float* __restrict__ b,
    const float* __restrict__ a1s, const float* __restrict__ a1b,
    float* __restrict__ h1)
{
  int m = blockIdx.x * 256 + threadIdx.x;
  if (m >= MN) return;
  float v[DD];
  float s = 0.f, q = 0.f;
  #pragma unroll
  for (int d = 0; d < DD; ++d) { float xv = t[(size_t)d*MN + m]; v[d] = xv; s += xv; q += xv*xv; }
  float mu  = s * (1.f/DD);
  float var = q * (1.f/DD) - mu*mu;        // biased (torch BN train)
  float rs  = rsqrtf(var + 1e-5f);
  float gm = g[m], bm = b[m];
  float s2 = 0.f, q2 = 0.f;
  #pragma unroll
  for (int d = 0; d < DD; ++d) {
    float y = lk((v[d]-mu)*rs*gm + bm);
    v[d] = y; s2 += y; q2 += y*y;
  }
  float mu2  = s2 * (1.f/DD);
  float var2 = (q2 - DD*mu2*mu2) * (1.f/(DD-1));   // unbiased (torch.std)
  float istd = 1.f/(sqrtf(fmaxf(var2,0.f)) + 1e-8f);
  #pragma unroll
  for (int d = 0; d < DD; ++d)
    h1[(size_t)m*DD + d] = a1s[d]*(v[d]-mu2)*istd + a1b[d];
}

// ------------- GEMM2: z[M][256] = h1[M][64] @ Wfc^T, + fused BN1 col sums -------
__global__ __launch_bounds__(256) void k_gemm2(const float* __restrict__ h1,
    const float* __restrict__ Wfc, float* __restrict__ z,
    float* __restrict__ colsum, float* __restrict__ colsumsq)
{
  const int tid  = threadIdx.x;
  const int wave = tid >> 5, lane = tid & 31;
  const int lo = lane & 15, hi = lane >> 4;
  const int mbase = blockIdx.x * 16;
  const int j0 = wave * 32, j1 = j0 + 16;
  v8f c0 = {}, c1 = {};
  for (int k = 0; k < DD; k += 4) {
    v2f a, b0, b1;
    float2 av  = *(const float2*)(h1  + (size_t)(mbase+lo)*DD + k + 2*hi);
    float2 bv0 = *(const float2*)(Wfc + (size_t)(j0  +lo)*DD + k + 2*hi);
    float2 bv1 = *(const float2*)(Wfc + (size_t)(j1  +lo)*DD + k + 2*hi);
    a.x=av.x;  a.y=av.y;  b0.x=bv0.x; b0.y=bv0.y; b1.x=bv1.x; b1.y=bv1.y;
    c0 = __builtin_amdgcn_wmma_f32_16x16x4_f32(false, a, false, b0, (short)0, c0, false, false);
    c1 = __builtin_amdgcn_wmma_f32_16x16x4_f32(false, a, false, b1, (short)0, c1, false, false);
  }
  float s0=0.f, q0=0.f, s1=0.f, q1=0.f;
  #pragma unroll
  for (int r = 0; r < 8; ++r) {
    int m = mbase + r + 8*hi;
    float v0 = c0[r], v1 = c1[r];
    z[(size_t)m*HH + j0 + lo] = v0;          // b_fc1 cancelled by BN
    z[(size_t)m*HH + j1 + lo] = v1;
    s0 += v0; q0 += v0*v0; s1 += v1; q1 += v1*v1;
  }
  atomicAdd(&colsum[j0+lo], s0); atomicAdd(&colsumsq[j0+lo], q0);
  atomicAdd(&colsum[j1+lo], s1); atomicAdd(&colsumsq[j1+lo], q1);
}

__global__ void k_bnfin(const float* __restrict__ cs, const float* __restrict__ csq,
                        float* __restrict__ mu, float* __restrict__ rs)
{
  int j = threadIdx.x;
  if (j < HH) {
    float m = cs[j] * (1.f/MN);
    float v = csq[j] * (1.f/MN) - m*m;
    mu[j] = m; rs[j] = rsqrtf(v + 1e-5f);
  }
}

// ------------- BN1 apply + leaky + adain2 (wave per row, wave32 shuffles) -------
__global__ __launch_bounds__(256) void k_applyC(const float* __restrict__ z,
    const float* __restrict__ mu, const float* __restrict__ rs,
    const float* __restrict__ g, const float* __restrict__ b,
    const float* __restrict__ a2s, const float* __restrict__ a2b,
    float* __restrict__ h2)
{
  int gid = blockIdx.x*256 + threadIdx.x;
  int m = gid >> 5, lane = gid & 31;
  if (m >= MN) return;
  const float* row = z + (size_t)m*HH;
  int j0 = lane*8;
  float4 v0 = *(const float4*)(row + j0);
  float4 v1 = *(const float4*)(row + j0 + 4);
  float y[8] = {v0.x,v0.y,v0.z,v0.w,v1.x,v1.y,v1.z,v1.w};
  float s = 0.f, q = 0.f;
  #pragma unroll
  for (int i = 0; i < 8; ++i) {
    int j = j0 + i;
    float yy = lk((y[i]-mu[j])*rs[j]*g[j] + b[j]);
    y[i] = yy; s += yy; q += yy*yy;
  }
  #pragma unroll
  for (int off = 16; off > 0; off >>= 1) { s += __shfl_xor(s, off, 32); q += __shfl_xor(q, off, 32); }
  float mur  = s * (1.f/HH);
  float var  = (q - HH*mur*mur) * (1.f/(HH-1));
  float istd = 1.f/(sqrtf(fmaxf(var,0.f)) + 1e-8f);
  #pragma unroll
  for (int i = 0; i < 8; ++i) {
    int j = j0 + i;
    h2[(size_t)m*HH + j] = a2s[j]*(y[i]-mur)*istd + a2b[j];
  }
}

// ------------- GEMM3: hg[M][64] = h2[M][256] @ Wg^T --------------------------
__global__ __launch_bounds__(128) void k_gemm3(const float* __restrict__ h2,
    const float* __restrict__ Wg, float* __restrict__ hg)
{
  const int tid  = threadIdx.x;
  const int wave = tid >> 5, lane = tid & 31;
  const int lo = lane & 15, hi = lane >> 4;
  const int mbase = blockIdx.x * 16;
  const int dbase = wave * 16;
  v8f c = {};
  for (int k = 0; k < HH; k += 4) {
    v2f a, b;
    float2 av = *(const float2*)(h2 + (size_t)(mbase+lo)*HH + k + 2*hi);
    float2 bv = *(const float2*)(Wg + (size_t)(dbase+lo)*HH + k + 2*hi);
    a.x=av.x; a.y=av.y; b.x=bv.x; b.y=bv.y;
    c = __builtin_amdgcn_wmma_f32_16x16x4_f32(false, a, false, b, (short)0, c, false, false);
  }
  #pragma unroll
  for (int r = 0; r < 8; ++r)
    hg[(size_t)(mbase + r + 8*hi)*DD + dbase + lo] = c[r];
}

// ------------- per-node attention logits ------------------------------------
__global__ __launch_bounds__(256) void k_attn(const float* __restrict__ hg,
    const float* __restrict__ asrc, const float* __restrict__ adst,
    float* __restrict__ as_, float* __restrict__ ad_)
{
  int m = blockIdx.x*256 + threadIdx.x;
  if (m >= MN) return;
  float s = 0.f, d = 0.f;
  #pragma unroll
  for (int i = 0; i < DD; i += 4) {
    float4 h = *(const float4*)(hg + (size_t)m*DD + i);
    s += h.x*asrc[i] + h.y*asrc[i+1] + h.z*asrc[i+2] + h.w*asrc[i+3];
    d += h.x*adst[i] + h.y*adst[i+1] + h.z*adst[i+2] + h.w*adst[i+3];
  }
  as_[m] = s; ad_[m] = d;
}

// ------------- GAT edge softmax (3 passes) + aggregation ---------------------
__global__ __launch_bounds__(256) void k_edge1(const int* __restrict__ ei,
    const float* __restrict__ as_, const float* __restrict__ ad_,
    float* __restrict__ ebuf, unsigned* __restrict__ maxk)
{
  int e = blockIdx.x*256 + threadIdx.x;
  if (e >= EE + MN) return;
  int s, d;
  if (e < EE) { s = ei[e]; d = ei[EE + e]; } else { s = d = e - EE; }
  float v = lk(as_[s] + ad_[d]);
  ebuf[e] = v;
  atomicMax(&maxk[d], fkey(v));
}

__global__ __launch_bounds__(256) void k_edge2(const int* __restrict__ ei,
    const unsigned* __restrict__ maxk, float* __restrict__ ebuf,
    float* __restrict__ denom)
{
  int e = blockIdx.x*256 + threadIdx.x;
  if (e >= EE + MN) return;
  int d = (e < EE) ? ei[EE + e] : (e - EE);
  float ex = expf(ebuf[e] - fdec(maxk[d]));
  ebuf[e] = ex;
  atomicAdd(&denom[d], ex);
}

__global__ __launch_bounds__(256) void k_edge3(const int* __restrict__ ei,
    const float* __restrict__ ebuf, const float* __restrict__ denom,
    const float* __restrict__ hg, float* __restrict__ out)
{
  int gid = blockIdx.x*256 + threadIdx.x;
  int e = gid >> 5, lane = gid & 31;
  if (e >= EE + MN) return;
  int s, d;
  if (e < EE) { s = ei[e]; d = ei[EE + e]; } else { s = d = e - EE; }
  float alpha = ebuf[e] / denom[d];
  float2 hv = *(const float2*)(hg + (size_t)s*DD + lane*2);
  atomicAdd(&out[(size_t)d*DD + lane*2    ], alpha*hv.x);
  atomicAdd(&out[(size_t)d*DD + lane*2 + 1], alpha*hv.y);
}

// ------------- + bg, leaky, adain3 (wave per 64-wide row) --------------------
__global__ __launch_bounds__(256) void k_final(const float* __restrict__ bg,
    const float* __restrict__ a3s, const float* __restrict__ a3b,
    float* __restrict__ out)
{
  int gid = blockIdx.x*256 + threadIdx.x;
  int m = gid >> 5, lane = gid & 31;
  if (m >= MN) return;
  int d0 = lane*2;
  float2 v = *(const float2*)(out + (size_t)m*DD + d0);
  float y0 = lk(v.x + bg[d0]), y1 = lk(v.y + bg[d0+1]);
  float s = y0 + y1, q = y0*y0 + y1*y1;
  #pragma unroll
  for (int off = 16; off > 0; off >>= 1) { s += __shfl_xor(s, off, 32); q += __shfl_xor(q, off, 32); }
  float mu   = s * (1.f/DD);
  float var  = (q - DD*mu*mu) * (1.f/(DD-1));
  float istd = 1.f/(sqrtf(fmaxf(var,0.f)) + 1e-8f);
  out[(size_t)m*DD + d0    ] = a3s[d0  ]*(y0-mu)*istd + a3b[d0  ];
  out[(size_t)m*DD + d0 + 1] = a3s[d0+1]*(y1-mu)*istd + a3b[d0+1];
}

// ------------- adain style-vector precompute ---------------------------------
__global__ void k_adainvec(const float* __restrict__ s1, const float* __restrict__ s2,
    const float* __restrict__ W1, const float* __restrict__ b1,
    const float* __restrict__ W2, const float* __restrict__ b2,
    const float* __restrict__ W3, const float* __restrict__ b3,
    const float* __restrict__ W4, const float* __restrict__ b4,
    int nfeat, float* __restrict__ scale, float* __restrict__ bias)
{
  int j = blockIdx.x*blockDim.x + threadIdx.x;
  if (j >= nfeat) return;
  float g1 = b1[j], be1 = b2[j], g2 = b3[j], be2 = b4[j];
  for (int i = 0; i < 512; ++i) {
    float sv = s1[i];
    g1  += sv * W1[(size_t)j*512 + i];
    be1 += sv * W2[(size_t)j*512 + i];
  }
  for (int i = 0; i < 64; ++i) {
    float sv = s2[i];
    g2  += sv * W3[(size_t)j*64 + i];
    be2 += sv * W4[(size_t)j*64 + i];
  }
  scale[j] = 0.5f*(g1+g2);
  bias[j]  = 0.5f*(be1+be2);
}

extern "C" void kernel_launch(void* const* d_in, const int* in_sizes, int n_in,
                              void* d_out, int out_size, void* d_ws, size_t ws_size,
                              hipStream_t stream)
{
  const float* x    = (const float*)d_in[0];
  const int*   ei   = (const int*)  d_in[1];
  const float* s1   = (const float*)d_in[2];
  const float* s2   = (const float*)d_in[3];
  const float* Wtrs = (const float*)d_in[4];
  const float* bn2g = (const float*)d_in[6];
  const float* bn2b = (const float*)d_in[7];
  const float* Wfc1 = (const float*)d_in[8];
  const float* bn1g = (const float*)d_in[10];
  const float* bn1b = (const float*)d_in[11];
  const float* Wg   = (const float*)d_in[12];
  const float* asrc = (const float*)d_in[13];
  const float* adst = (const float*)d_in[14];
  const float* bg   = (const float*)d_in[15];
  const float *a1W1=(const float*)d_in[16], *a1b1=(const float*)d_in[17];
  const float *a1W2=(const float*)d_in[18], *a1b2=(const float*)d_in[19];
  const float *a1W3=(const float*)d_in[20], *a1b3=(const float*)d_in[21];
  const float *a1W4=(const float*)d_in[22], *a1b4=(const float*)d_in[23];
  const float *a2W1=(const float*)d_in[24], *a2b1=(const float*)d_in[25];
  const float *a2W2=(const float*)d_in[26], *a2b2=(const float*)d_in[27];
  const float *a2W3=(const float*)d_in[28], *a2b3=(const float*)d_in[29];
  const float *a2W4=(const float*)d_in[30], *a2b4=(const float*)d_in[31];
  const float *a3W1=(const float*)d_in[32], *a3b1=(const float*)d_in[33];
  const float *a3W2=(const float*)d_in[34], *a3b2=(const float*)d_in[35];
  const float *a3W3=(const float*)d_in[36], *a3b3=(const float*)d_in[37];
  const float *a3W4=(const float*)d_in[38], *a3b4=(const float*)d_in[39];

  // workspace layout (floats): total ~7.42M floats (~29.7 MB)
  float* ws  = (float*)d_ws;
  float* t   = ws + 0;              // 640000
  float* h1  = ws + 640000;         // 640000
  float* z   = ws + 1280000;        // 2560000
  float* h2  = ws + 3840000;        // 2560000
  float* hg  = ws + 6400000;        // 640000
  float* as_ = ws + 7040000;        // 10000
  float* ad_ = ws + 7050000;        // 10000
  unsigned* maxk = (unsigned*)(ws + 7060000); // 10000
  float* den  = ws + 7070000;       // 10000
  float* ebuf = ws + 7080000;       // 330000
  float* cs   = ws + 7410000;       // 256
  float* csq  = ws + 7410256;       // 256
  float* muv  = ws + 7410512;       // 256
  float* rsv  = ws + 7410768;       // 256
  float* A1S  = ws + 7411024; float* A1B = ws + 7411088;
  float* A2S  = ws + 7411152; float* A2B = ws + 7411408;
  float* A3S  = ws + 7411664; float* A3B = ws + 7411728;

  float* out = (float*)d_out;
  const int ne = EE + MN;

  // zero accumulators every call (ws/out are poisoned, never re-zeroed by harness)
  (void)hipMemsetAsync(cs,   0, 512 * sizeof(float), stream);   // cs + csq contiguous
  (void)hipMemsetAsync(den,  0, MN  * sizeof(float), stream);
  (void)hipMemsetAsync(maxk, 0, MN  * sizeof(unsigned), stream);
  (void)hipMemsetAsync(out,  0, (size_t)MN * DD * sizeof(float), stream);

  k_adainvec<<<1,  64, 0, stream>>>(s1, s2, a1W1,a1b1,a1W2,a1b2,a1W3,a1b3,a1W4,a1b4,  64, A1S, A1B);
  k_adainvec<<<1, 256, 0, stream>>>(s1, s2, a2W1,a2b1,a2W2,a2b2,a2W3,a2b3,a2W4,a2b4, 256, A2S, A2B);
  k_adainvec<<<1,  64, 0, stream>>>(s1, s2, a3W1,a3b1,a3W2,a3b2,a3W3,a3b3,a3W4,a3b4,  64, A3S, A3B);

  k_gemm1 <<<625, 128, 0, stream>>>(x, Wtrs, t);
  k_stageB<<<(MN+255)/256, 256, 0, stream>>>(t, bn2g, bn2b, A1S, A1B, h1);
  k_gemm2 <<<625, 256, 0, stream>>>(h1, Wfc1, z, cs, csq);
  k_bnfin <<<1, 256, 0, stream>>>(cs, csq, muv, rsv);
  k_applyC<<<(MN*32+255)/256, 256, 0, stream>>>(z, muv, rsv, bn1g, bn1b, A2S, A2B, h2);
  k_gemm3 <<<625, 128, 0, stream>>>(h2, Wg, hg);
  k_attn  <<<(MN+255)/256, 256, 0, stream>>>(hg, asrc, adst, as_, ad_);
  k_edge1 <<<(ne+255)/256, 256, 0, stream>>>(ei, as_, ad_, ebuf, maxk);
  k_edge2 <<<(ne+255)/256, 256, 0, stream>>>(ei, maxk, ebuf, den);
  k_edge3 <<<(ne*32+255)/256, 256, 0, stream>>>(ei, ebuf, den, hg, out);
  k_final <<<(MN*32+255)/256, 256, 0, stream>>>(bg, A3S, A3B, out);
}